// HeteroDotProductPredictor_1047972020372
// MI455X (gfx1250) — compile-verified
//
#include <hip/hip_runtime.h>

typedef float v2f __attribute__((ext_vector_type(2)));
typedef float v8f __attribute__((ext_vector_type(8)));

// One wave32 handles a tile of 16 edges.
// Scores = diag( S(16x256) x Dm(16x256)^T ), computed as 64 chained
// V_WMMA_F32_16X16X4_F32 steps (exact fp32 math).
//
// f32 WMMA operand layout (16x16x4):
//   A (16x4, M x K): lane m      holds {A[m][k+0], A[m][k+1]}
//                    lane m+16   holds {A[m][k+2], A[m][k+3]}
//   B (4x16, K x N): lane n      holds {B[0][n],  B[1][n]}   = {Dm[n][k+0], Dm[n][k+1]}
//                    lane n+16   holds {B[2][n],  B[3][n]}   = {Dm[n][k+2], Dm[n][k+3]}
// => both operands use the SAME per-lane addressing: row (lane&15),
//    columns 4t + 2*(lane>>4) + {0,1}.  Two global_load_b64 per step.
__global__ __launch_bounds__(256) void edge_dot_wmma_f32(
    const float* __restrict__ h,   // [N, 256] fp32 node features
    const int*   __restrict__ src, // [E]
    const int*   __restrict__ dst, // [E]
    float*       __restrict__ out, // [E] scores
    int nEdges, int nGroups)
{
    const int wave = threadIdx.x >> 5;
    const int lane = threadIdx.x & 31;
    const int group = blockIdx.x * 8 + wave;
    if (group >= nGroups) return;          // wave-uniform exit, EXEC stays full

    const int e0   = group << 4;           // first edge of this 16-edge tile
    const int row  = lane & 15;            // edge-in-tile this lane feeds
    const int coff = (lane >> 4) << 1;     // column sub-offset: 0 or 2

    int e = e0 + row;                      // clamp tail instead of masking:
    if (e >= nEdges) e = nEdges - 1;       // WMMA requires EXEC == all ones

    const float* ps = h + (size_t)src[e] * 256u + coff;   // A source (src row)
    const float* pd = h + (size_t)dst[e] * 256u + coff;   // B source (dst row)

    v8f c0 = {};
    v8f c1 = {};
    #pragma unroll 16
    for (int t = 0; t < 64; t += 2) {
        v2f a0 = *(const v2f*)(ps + 4 * t);
        v2f b0 = *(const v2f*)(pd + 4 * t);
        c0 = __builtin_amdgcn_wmma_f32_16x16x4_f32(
                 false, a0, false, b0, (short)0, c0, false, false);
        v2f a1 = *(const v2f*)(ps + 4 * t + 4);
        v2f b1 = *(const v2f*)(pd + 4 * t + 4);
        c1 = __builtin_amdgcn_wmma_f32_16x16x4_f32(
                 false, a1, false, b1, (short)0, c1, false, false);
    }
    v8f c = c0 + c1;

    // C/D layout: lane n (<16), VGPR v holds D[v][n]; lane n+16 holds D[v+8][n].
    // Diagonal D[i][i]: lanes 0..7 -> VGPR = lane; lanes 24..31 -> VGPR = lane-24.
    const int q = lane & 7;
    float v = c[0];
    v = (q == 1) ? c[1] : v;
    v = (q == 2) ? c[2] : v;
    v = (q == 3) ? c[3] : v;
    v = (q == 4) ? c[4] : v;
    v = (q == 5) ? c[5] : v;
    v = (q == 6) ? c[6] : v;
    v = (q == 7) ? c[7] : v;

    const bool writer = (lane < 8) || (lane >= 24);
    const int  m      = (lane < 8) ? lane : (lane - 16);   // diag index 0..15
    const int  eo     = e0 + m;
    if (writer && eo < nEdges) out[eo] = v;
}

extern "C" void kernel_launch(void* const* d_in, const int* in_sizes, int n_in,
                              void* d_out, int out_size, void* d_ws, size_t ws_size,
                              hipStream_t stream) {
    (void)n_in; (void)out_size; (void)d_ws; (void)ws_size;
    const float* h   = (const float*)d_in[0];
    const int*   src = (const int*)d_in[1];
    const int*   dst = (const int*)d_in[2];
    float*       out = (float*)d_out;

    const int nEdges  = in_sizes[1];            // E
    const int nGroups = (nEdges + 15) >> 4;     // 16-edge tiles
    const int blocks  = (nGroups + 7) >> 3;     // 8 waves (tiles) per block

    edge_dot_wmma_f32<<<blocks, 256, 0, stream>>>(h, src, dst, out, nEdges, nGroups);
}